// GaussianSplatting2D_13288628814348
// MI455X (gfx1250) — compile-verified
//
#include <hip/hip_runtime.h>
#include <hip/hip_bf16.h>
#include <math.h>

// ---------------------------------------------------------------------------
// 2D Gaussian splatting, W=H=256, N=1000, C=1.  Compute-bound: 65.5M
// gaussian-pixel evals, negligible memory traffic at 23.3 TB/s.
//
// Exponent e(g,p) = log2(sigmoid(o)) - (q/2)*log2(e) is a rank-6 bilinear
// form evaluated per 16x16 (gaussian x pixel) tile with two chained
// V_WMMA_F32_16X16X4_F32 ops.  All coefficients are pre-scaled by log2(e) in
// the prep kernel so the WMMA output feeds v_exp_f32 (exp2) directly with no
// per-element scaling.  The alpha clamp is applied in the log2 domain.
//
// Serial compositing: each lane owns 1 pixel and 8 consecutive gaussians of
// the block; one shfl_xor(16) stitches the half-blocks in order.  Blocks
// whose 16 gaussians are all negligible for this tile (lane-max exponent
// < -30, i.e. alpha < 1e-9) are skipped with a wave-uniform branch, so EXEC
// stays all-ones around the WMMAs.
// ---------------------------------------------------------------------------

typedef float v2f __attribute__((ext_vector_type(2)));
typedef float v8f __attribute__((ext_vector_type(8)));

#define GS_W 256
#define GS_LOG2E 1.4426950408889634f
#define GS_LOG2_AMAX -0.0014434690f   /* log2(0.999) */
#define GS_SKIP_THRESH -30.0f         /* exp2(-30) ~ 1e-9 */

// ---------------------------------------------------------------------------
// Prep: per-gaussian coefficients (padded to a multiple of 16), log2-domain.
//   params[2g+0] = { -0.5*ia*L2E, -ib*L2E, -0.5*ic*L2E, mx }
//   params[2g+1] = { my, log(sigmoid(opac))*L2E, 0, 0 }
//   cols[g]      = sigmoid(rgb)
// Pad gaussians get logopac = -1e30 -> alpha == 0 -> inert / culled.
// ---------------------------------------------------------------------------
__global__ void gs2d_prep(const float* __restrict__ means,
                          const float* __restrict__ quats,
                          const float* __restrict__ scales,
                          const float* __restrict__ rgbs,
                          const float* __restrict__ opacities,
                          float4* __restrict__ params,
                          float* __restrict__ cols,
                          int n, int npad) {
    int g = blockIdx.x * blockDim.x + threadIdx.x;
    if (g >= npad) return;
    if (g >= n) {
        params[2 * g + 0] = make_float4(0.f, 0.f, 0.f, 0.f);
        params[2 * g + 1] = make_float4(0.f, -1e30f, 0.f, 0.f);
        cols[g] = 0.f;
        return;
    }
    float mx = means[2 * g + 0];
    float my = means[2 * g + 1];
    float qa = quats[g];
    float sx = scales[2 * g + 0];
    float sy = scales[2 * g + 1];
    float c = cosf(qa), s = sinf(qa);
    float sx2 = sx * sx, sy2 = sy * sy;
    float a11 = c * c * sx2 + s * s * sy2;
    float a12 = c * s * (sx2 - sy2);
    float a22 = s * s * sx2 + c * c * sy2;
    float det = a11 * a22 - a12 * a12;
    float inv = 1.0f / det;
    float ia = a22 * inv;
    float ib = -a12 * inv;
    float ic = a11 * inv;
    // log(sigmoid(o)), stable
    float o = opacities[g];
    float lop = (o > 0.f) ? -log1pf(expf(-o)) : (o - log1pf(expf(o)));
    params[2 * g + 0] = make_float4(-0.5f * GS_LOG2E * ia,
                                    -GS_LOG2E * ib,
                                    -0.5f * GS_LOG2E * ic, mx);
    params[2 * g + 1] = make_float4(my, GS_LOG2E * lop, 0.f, 0.f);
    float r = rgbs[g];  // C == 1
    cols[g] = 1.0f / (1.0f + expf(-r));
}

// ---------------------------------------------------------------------------
// Render: one wave per 4x4 pixel tile (64x64 tiles = 4096 waves).
// WMMA: M = 16 gaussians, K = features, N = 16 pixels.
//   e = q0*u^2 + q1*u*v + q2*v^2 + q3*u + q4*v + q5   (log2 domain)
// Features split over two K=4 WMMAs: [u^2, uv, v^2, u] then [v, 1, 0, 0].
// f32 16x4 A layout: lanes 0-15 hold M=lane, K=0,1; lanes 16-31 hold K=2,3.
// B 4x16 mirrors it.  D: lane = pixel N=lane%16, rows M = r + 8*(lane>=16).
// ---------------------------------------------------------------------------
__global__ __launch_bounds__(256) void gs2d_render(
        const float4* __restrict__ params,
        const float* __restrict__ cols,
        float* __restrict__ out,
        int nblk) {
    int tid = blockIdx.x * blockDim.x + threadIdx.x;
    int wave = tid >> 5;
    int lane = threadIdx.x & 31;
    bool upper = lane >= 16;
    int lp = lane & 15;

    // 4x4 pixel tile; tile-centered coords keep all features O(1) in f32.
    int tx = wave & 63;
    int ty = wave >> 6;
    float xc = (float)(tx * 4) + 2.0f;
    float yc = (float)(ty * 4) + 2.0f;
    int pux = lp & 3;
    int puy = lp >> 2;
    float u = (float)pux - 1.5f;  // x - xc
    float v = (float)puy - 1.5f;  // y - yc

    // B matrices (pixel features), loop-invariant.
    v2f b1, b2;
    b1.x = upper ? (v * v) : (u * u);  // K2 : K0
    b1.y = upper ? u       : (u * v);  // K3 : K1
    b2.x = upper ? 0.f : v;
    b2.y = upper ? 0.f : 1.f;

    float acc = 0.0f;  // accumulated color (C == 1)
    float T   = 1.0f;  // running transmittance

    for (int blk = 0; blk < nblk; ++blk) {
        int base = blk << 4;

        // Tile-local coefficients for gaussian g = base + lp (factored form,
        // no cancellation; everything already scaled by log2(e)).
        float4 p0 = params[2 * (base + lp) + 0];
        float4 p1 = params[2 * (base + lp) + 1];
        float q0 = p0.x, q1 = p0.y, q2 = p0.z;
        float dxc = xc - p0.w;
        float dyc = yc - p1.x;
        float q3 = q0 * (dxc + dxc) + q1 * dyc;
        float q4 = q1 * dxc + q2 * (dyc + dyc);
        float q5 = (q0 * dxc * dxc + q1 * dxc * dyc + q2 * dyc * dyc) + p1.y;

        v2f a1, a2;
        a1.x = upper ? q2 : q0;   // K2 : K0
        a1.y = upper ? q3 : q1;   // K3 : K1
        a2.x = upper ? 0.f : q4;
        a2.y = upper ? 0.f : q5;

        v8f d = {};
        d = __builtin_amdgcn_wmma_f32_16x16x4_f32(false, a1, false, b1,
                                                  (short)0, d, false, false);
        d = __builtin_amdgcn_wmma_f32_16x16x4_f32(false, a2, false, b2,
                                                  (short)0, d, false, false);

        // Wave-uniform block culling: skip composite if every exponent in the
        // tile is below threshold (alpha < 1e-9).  Scalar branch -> EXEC
        // untouched; T/acc unchanged is exact (s=0, P=1 identity).
        float m = d[0];
#pragma unroll
        for (int r = 1; r < 8; ++r) m = fmaxf(m, d[r]);
        if (__builtin_amdgcn_ballot_w32(m > GS_SKIP_THRESH) == 0u) continue;

        // Colors for this lane's 8 gaussians (base + 8*upper + 0..7).
        const float4* c4 = (const float4*)(cols + base + (upper ? 8 : 0));
        float4 cA = c4[0];
        float4 cB = c4[1];
        float cr[8] = {cA.x, cA.y, cA.z, cA.w, cB.x, cB.y, cB.z, cB.w};

        // Local front-to-back composite over 8 consecutive gaussians.
        // alpha = min(exp2(d), 0.999) applied in the log2 domain.
        float s = 0.0f, P = 1.0f;
#pragma unroll
        for (int r = 0; r < 8; ++r) {
            float a = __builtin_amdgcn_exp2f(fminf(d[r], GS_LOG2_AMAX));
            s += P * (a * cr[r]);
            P *= (1.0f - a);
        }

        // Stitch the two half-blocks in gaussian order.
        float so = __shfl_xor(s, 16, 32);
        float Po = __shfl_xor(P, 16, 32);
        float s_lo = upper ? so : s;
        float P_lo = upper ? Po : P;
        float s_hi = upper ? s : so;
        float P_hi = upper ? P : Po;

        acc += T * (s_lo + P_lo * s_hi);
        T *= P_lo * P_hi;
    }

    if (!upper) {
        int x = tx * 4 + pux;
        int y = ty * 4 + puy;
        out[y * GS_W + x] = acc;  // output shape (1,1,256,256)
    }
}

extern "C" void kernel_launch(void* const* d_in, const int* in_sizes, int n_in,
                              void* d_out, int out_size, void* d_ws, size_t ws_size,
                              hipStream_t stream) {
    const float* means     = (const float*)d_in[0];
    const float* quats     = (const float*)d_in[1];
    const float* scales    = (const float*)d_in[2];
    const float* rgbs      = (const float*)d_in[3];
    const float* opacities = (const float*)d_in[4];

    int n    = in_sizes[1];            // N from quats (N,)
    int npad = (n + 15) & ~15;         // pad to block of 16
    int nblk = npad >> 4;

    // Workspace layout: params (2 float4 per gaussian) then colors.
    float4* params = (float4*)d_ws;
    float*  colsw  = (float*)((char*)d_ws +
                     (((size_t)npad * 2 * sizeof(float4) + 255) & ~(size_t)255));

    gs2d_prep<<<(npad + 255) / 256, 256, 0, stream>>>(
        means, quats, scales, rgbs, opacities, params, colsw, n, npad);

    // 4096 tiles (64x64 of 4x4 pixels), 1 wave each -> 512 blocks of 8 waves.
    gs2d_render<<<512, 256, 0, stream>>>(params, colsw, (float*)d_out, nblk);
}